// PointNetPlusPlus_34943853920282
// MI455X (gfx1250) — compile-verified
//
#include <hip/hip_runtime.h>
#include <hip/hip_bf16.h>

// ---------------------------------------------------------------------------
// PointNet++ forward for MI455X (gfx1250, wave32).
// All MLP layers + head run through V_WMMA_F32_16X16X4_F32 (fp32 matches the
// reference numerics; the problem is L2-resident and latency-bound, not
// bandwidth-bound, so fp32 WMMA is the right precision path).
// ---------------------------------------------------------------------------

typedef __attribute__((ext_vector_type(2))) float v2f;
typedef __attribute__((ext_vector_type(8))) float v8f;

#define BATCH 8
#define NPTS 4096
#define NSAMPLE 16

#define WMMA_F32(a, b, c) \
  __builtin_amdgcn_wmma_f32_16x16x4_f32(false, (a), false, (b), (short)0, (c), \
                                        false, false)

// ---------------------------- FPS (block per batch) -------------------------
__global__ void fps_kernel(const float* __restrict__ pts, int N, int C, int K,
                           float* __restrict__ sel, int* __restrict__ idx) {
  int b = blockIdx.x;
  const float* P = pts + (size_t)b * N * C;
  float* S = sel + (size_t)b * K * C;
  int* I = idx + b * K;
  extern __shared__ float smem[];
  float* mind = smem;              // [N]
  float* lastp = smem + N;         // [C]
  float* redv = lastp + C;         // [256]
  int* redi = (int*)(redv + 256);  // [256]
  int t = threadIdx.x;

  for (int n = t; n < N; n += blockDim.x) mind[n] = 1e10f;
  if (t == 0) I[0] = 0;
  for (int c = t; c < C; c += blockDim.x) S[c] = P[c];
  int last = 0;
  __syncthreads();

  for (int k = 1; k < K; ++k) {
    for (int c = t; c < C; c += blockDim.x) lastp[c] = P[(size_t)last * C + c];
    __syncthreads();
    float bestv = -1.0f;
    int besti = 0;
    for (int n = t; n < N; n += blockDim.x) {
      const float* p = P + (size_t)n * C;
      float d = 0.f;
      for (int c = 0; c < C; ++c) {
        float df = p[c] - lastp[c];
        d += df * df;
      }
      float m = fminf(mind[n], d);
      mind[n] = m;
      if (m > bestv) { bestv = m; besti = n; }
    }
    redv[t] = bestv;
    redi[t] = besti;
    __syncthreads();
    for (int s = 128; s > 0; s >>= 1) {
      if (t < s) {
        if (redv[t + s] > redv[t] ||
            (redv[t + s] == redv[t] && redi[t + s] < redi[t])) {
          redv[t] = redv[t + s];
          redi[t] = redi[t + s];
        }
      }
      __syncthreads();
    }
    last = redi[0];
    if (t == 0) I[k] = last;
    for (int c = t; c < C; c += blockDim.x)
      S[(size_t)k * C + c] = P[(size_t)last * C + c];
    __syncthreads();
  }
}

// ----------------------- ball query: first-16 within radius -----------------
__global__ void ball_query_kernel(const float* __restrict__ sel,
                                  const float* __restrict__ pts, int N, int C,
                                  int K, float r2, float* __restrict__ grouped,
                                  int nq) {
  int q = blockIdx.x * blockDim.x + threadIdx.x;
  if (q >= nq) return;
  int b = q / K;
  const float* Q = sel + (size_t)q * C;
  const float* P = pts + (size_t)b * N * C;
  float* G = grouped + (size_t)q * NSAMPLE * C;
  int cnt = 0;
  for (int n = 0; n < N && cnt < NSAMPLE; ++n) {
    const float* p = P + (size_t)n * C;
    float d = 0.f;
    for (int c = 0; c < C; ++c) {
      float df = Q[c] - p[c];
      d += df * df;
    }
    if (d < r2) {
      for (int c = 0; c < C; ++c) G[cnt * C + c] = p[c];
      ++cnt;
    }
  }
  for (; cnt < NSAMPLE; ++cnt)
    for (int c = 0; c < C; ++c) G[cnt * C + c] = 0.f;
}

// --------------------- WMMA fp32 GEMM: Y = X*W^T + bias ---------------------
// X: (R, C) rows with stride ldx, column offset xoff. W: (O, C). Y: (R, O).
// Each wave owns a 32x16 output block (two 16x16 M-tiles sharing the B
// fragment). K stepped by 4 via V_WMMA_F32_16X16X4_F32. The main loop is
// bounds-check-free and software-pipelined (loads for chunk i+1 issued before
// the WMMAs of chunk i), so waits become partial instead of loadcnt==0.
__global__ void __launch_bounds__(128) gemm_wmma_kernel(
    const float* __restrict__ X, int ldx, int xoff, const float* __restrict__ W,
    const float* __restrict__ bias, float* __restrict__ Y, int R, int C,
    int O) {
  int wave = (blockIdx.x * blockDim.x + threadIdx.x) >> 5;
  int lane = threadIdx.x & 31;
  int tilesN = O >> 4;
  int tilesM = R >> 5;  // 32 rows per wave
  int tm = wave / tilesN;
  int tn = wave - tm * tilesN;
  if (tm >= tilesM) return;  // wave-uniform: EXEC stays all-ones for WMMA
  int r0 = tm << 5, n0 = tn << 4;
  int half = lane >> 4;  // 0: K pair {0,1}, 1: K pair {2,3}
  int l15 = lane & 15;
  int koff = half * 2;

  const float* A0 = X + (size_t)(r0 + l15) * ldx + xoff + koff;
  const float* A1 = A0 + (size_t)16 * ldx;
  const float* Bp = W + (size_t)(n0 + l15) * C + koff;
  v8f acc0 = {};
  v8f acc1 = {};
  int C4 = C & ~3;
  int c0 = 0;
  if (C4 >= 4) {
    // prologue: load chunk 0
    v2f a0c, a1c, bbc;
    a0c.x = A0[0];
    a0c.y = A0[1];
    a1c.x = A1[0];
    a1c.y = A1[1];
    bbc.x = Bp[0];
    bbc.y = Bp[1];
    // steady state: prefetch chunk i+1, then WMMA chunk i
#pragma unroll 4
    for (c0 = 0; c0 + 8 <= C4; c0 += 4) {
      int kk = c0 + 4;
      v2f a0n, a1n, bbn;
      a0n.x = A0[kk];
      a0n.y = A0[kk + 1];
      a1n.x = A1[kk];
      a1n.y = A1[kk + 1];
      bbn.x = Bp[kk];
      bbn.y = Bp[kk + 1];
      acc0 = WMMA_F32(a0c, bbc, acc0);
      acc1 = WMMA_F32(a1c, bbc, acc1);
      a0c = a0n;
      a1c = a1n;
      bbc = bbn;
    }
    // epilogue: last full chunk
    acc0 = WMMA_F32(a0c, bbc, acc0);
    acc1 = WMMA_F32(a1c, bbc, acc1);
    c0 += 4;
  }
  if (c0 < C) {  // ragged-K tail, executed once; EXEC restored before WMMA
    int k0 = c0 + koff;
    v2f a0 = {0.f, 0.f}, a1 = {0.f, 0.f}, bb = {0.f, 0.f};
    if (k0 < C) {
      a0.x = A0[c0];
      a1.x = A1[c0];
      bb.x = Bp[c0];
    }
    if (k0 + 1 < C) {
      a0.y = A0[c0 + 1];
      a1.y = A1[c0 + 1];
      bb.y = Bp[c0 + 1];
    }
    acc0 = WMMA_F32(a0, bb, acc0);
    acc1 = WMMA_F32(a1, bb, acc1);
  }
  int col = n0 + l15;
  float bv = bias ? bias[col] : 0.f;
  int rbase = r0 + (half ? 8 : 0);
#pragma unroll
  for (int r = 0; r < 8; ++r)
    Y[(size_t)(rbase + r) * O + col] = acc0[r] + bv;
#pragma unroll
  for (int r = 0; r < 8; ++r)
    Y[(size_t)(rbase + 16 + r) * O + col] = acc1[r] + bv;
}

// ------------- BN stats over all rows per channel -> scale/shift ------------
__global__ void bn_stats_kernel(const float* __restrict__ Y, int R, int O,
                                const float* __restrict__ g,
                                const float* __restrict__ be,
                                float* __restrict__ scale,
                                float* __restrict__ shift) {
  int o = blockIdx.x;
  __shared__ float s1[256], s2[256];
  float a = 0.f, b = 0.f;
  for (int r = threadIdx.x; r < R; r += blockDim.x) {
    float v = Y[(size_t)r * O + o];
    a += v;
    b += v * v;
  }
  s1[threadIdx.x] = a;
  s2[threadIdx.x] = b;
  __syncthreads();
  for (int s = 128; s > 0; s >>= 1) {
    if (threadIdx.x < s) {
      s1[threadIdx.x] += s1[threadIdx.x + s];
      s2[threadIdx.x] += s2[threadIdx.x + s];
    }
    __syncthreads();
  }
  if (threadIdx.x == 0) {
    float inv = 1.0f / (float)R;
    float mu = s1[0] * inv;
    float var = s2[0] * inv - mu * mu;
    float sc = g[o] * rsqrtf(var + 1e-5f);
    scale[o] = sc;
    shift[o] = be[o] - mu * sc;
  }
}

__global__ void bn_relu_kernel(float* __restrict__ Y,
                               const float* __restrict__ scale,
                               const float* __restrict__ shift, size_t n,
                               int O) {
  size_t i = (size_t)blockIdx.x * blockDim.x + threadIdx.x;
  if (i >= n) return;
  int o = (int)(i % O);
  float v = Y[i] * scale[o] + shift[o];
  Y[i] = v > 0.f ? v : 0.f;
}

// ----------------------- max over nsample dimension -------------------------
__global__ void maxpool_kernel(const float* __restrict__ X,
                               float* __restrict__ out, int BM, int S, int O) {
  int i = blockIdx.x * blockDim.x + threadIdx.x;
  if (i >= BM * O) return;
  int bm = i / O, o = i - bm * O;
  const float* p = X + ((size_t)bm * S) * O + o;
  float m = p[0];
  for (int s = 1; s < S; ++s) m = fmaxf(m, p[(size_t)s * O]);
  out[i] = m;
}

// ------------------------ concat [xyz(3) , feat(O)] -------------------------
__global__ void concat_kernel(const float* __restrict__ xyz, int ldxyz,
                              const float* __restrict__ feat, int O,
                              float* __restrict__ out, size_t total) {
  size_t i = (size_t)blockIdx.x * blockDim.x + threadIdx.x;
  if (i >= total) return;
  int Cout = 3 + O;
  size_t r = i / Cout;
  int c = (int)(i - r * Cout);
  out[i] = (c < 3) ? xyz[r * ldxyz + c] : feat[r * O + (c - 3)];
}

// --------------- 3-NN inverse-distance interpolation (per point) ------------
__global__ void interp_kernel(const float* __restrict__ p1, int C1, int N1,
                              const float* __restrict__ p2, int C2, int N2,
                              float* __restrict__ ip, int nq) {
  int q = blockIdx.x * blockDim.x + threadIdx.x;
  if (q >= nq) return;
  int b = q / N1;
  const float* P1 = p1 + (size_t)q * C1;
  const float* P2 = p2 + (size_t)b * N2 * C2;
  float d0 = 1e30f, d1 = 1e30f, dd2 = 1e30f;
  int i0 = 0, i1 = 0, i2 = 0;
  for (int m = 0; m < N2; ++m) {
    const float* p = P2 + (size_t)m * C2;
    float dx = P1[0] - p[0], dy = P1[1] - p[1], dz = P1[2] - p[2];
    float d = dx * dx + dy * dy + dz * dz;
    if (d < d0) {
      dd2 = d1; i2 = i1; d1 = d0; i1 = i0; d0 = d; i0 = m;
    } else if (d < d1) {
      dd2 = d1; i2 = i1; d1 = d; i1 = m;
    } else if (d < dd2) {
      dd2 = d; i2 = m;
    }
  }
  float w0 = 1.f / (sqrtf(fmaxf(d0, 0.f)) + 1e-8f);
  float w1 = 1.f / (sqrtf(fmaxf(d1, 0.f)) + 1e-8f);
  float w2 = 1.f / (sqrtf(fmaxf(dd2, 0.f)) + 1e-8f);
  float ws = w0 + w1 + w2;
  w0 /= ws; w1 /= ws; w2 /= ws;
  int F2 = C2 - 3;
  float* out = ip + (size_t)q * (C1 + F2);
  for (int c = 0; c < C1; ++c) out[c] = P1[c];
  const float* f0 = P2 + (size_t)i0 * C2 + 3;
  const float* f1 = P2 + (size_t)i1 * C2 + 3;
  const float* f2 = P2 + (size_t)i2 * C2 + 3;
  for (int f = 0; f < F2; ++f)
    out[C1 + f] = w0 * f0[f] + w1 * f1[f] + w2 * f2[f];
}

__global__ void copy_kernel(const float* __restrict__ src,
                            float* __restrict__ dst, size_t n) {
  size_t i = (size_t)blockIdx.x * blockDim.x + threadIdx.x;
  if (i < n) dst[i] = src[i];
}

// ---------------------------------------------------------------------------
// Host orchestration
// ---------------------------------------------------------------------------

// workspace arena (float offsets); total ~107.7 MB
static constexpr size_t O_SEL = 0;         // 140000
static constexpr size_t O_IDX = 140000;    // 8192 (as int)
static constexpr size_t O_G = 150000;      // 2300000
static constexpr size_t O_T1 = 2450000;    // 8400000
static constexpr size_t O_T2 = 10850000;   // 4200000
static constexpr size_t O_POOL = 15050000; // 530000
static constexpr size_t O_SC = 15581000;   // 512
static constexpr size_t O_SH = 15582000;   // 512
static constexpr size_t O_S1 = 15583000;   // 548864
static constexpr size_t O_S2 = 16132000;   // 268288
static constexpr size_t O_S3 = 16401000;   // 132608
static constexpr size_t O_S4 = 16534000;   // 65920
static constexpr size_t O_IP = 16600000;   // 4300000
static constexpr size_t O_F1 = 20900000;   // 132608
static constexpr size_t O_F2 = 21033000;   // 530432
static constexpr size_t O_F3 = 21564000;   // 1073152
static constexpr size_t O_F4 = 22638000;   // 4292608

static void launch_gemm(hipStream_t stream, const float* X, int ldx, int xoff,
                        const float* W, const float* b, float* Y, int rows,
                        int C, int O) {
  int tiles = (rows / 32) * (O / 16);
  int blocks = (tiles + 3) / 4;  // 4 waves / 128-thread block
  gemm_wmma_kernel<<<blocks, 128, 0, stream>>>(X, ldx, xoff, W, b, Y, rows, C,
                                               O);
}

static void pointnet_run(hipStream_t stream, void* const* d_in, int pbase,
                         int nlayers, const int* outs, const float* X0, int ldx,
                         int xoff, int rows, int inC, float* T1, float* T2,
                         float* scaleB, float* shiftB, const float** finalOut,
                         int* finalO) {
  const float* X = X0;
  int ldcur = ldx, off = xoff, C = inC;
  float* bufs[2] = {T1, T2};
  int cur = 0;
  for (int l = 0; l < nlayers; ++l) {
    const float* W = (const float*)d_in[pbase + 4 * l + 0];
    const float* b = (const float*)d_in[pbase + 4 * l + 1];
    const float* g = (const float*)d_in[pbase + 4 * l + 2];
    const float* be = (const float*)d_in[pbase + 4 * l + 3];
    int O = outs[l];
    float* Y = bufs[cur];
    launch_gemm(stream, X, ldcur, off, W, b, Y, rows, C, O);
    bn_stats_kernel<<<O, 256, 0, stream>>>(Y, rows, O, g, be, scaleB, shiftB);
    size_t n = (size_t)rows * O;
    bn_relu_kernel<<<(int)((n + 255) / 256), 256, 0, stream>>>(Y, scaleB,
                                                               shiftB, n, O);
    X = Y; ldcur = O; off = 0; C = O;
    cur ^= 1;
  }
  *finalOut = X;
  *finalO = C;
}

static void sa_stage(hipStream_t stream, void* const* d_in, const float* P,
                     int N, int Cin, int K, float radius, int pbase,
                     const int* outs, int nl, float* ws, float* stageOut) {
  float* sel = ws + O_SEL;
  int* idxb = (int*)(ws + O_IDX);
  float* G = ws + O_G;
  int shmem = (N + Cin + 512) * (int)sizeof(float);
  fps_kernel<<<BATCH, 256, shmem, stream>>>(P, N, Cin, K, sel, idxb);
  int nq = BATCH * K;
  ball_query_kernel<<<(nq + 127) / 128, 128, 0, stream>>>(
      sel, P, N, Cin, K, radius * radius, G, nq);
  int rows = nq * NSAMPLE;
  const float* feat;
  int O;
  pointnet_run(stream, d_in, pbase, nl, outs, G, Cin, 0, rows, Cin, ws + O_T1,
               ws + O_T2, ws + O_SC, ws + O_SH, &feat, &O);
  int npool = nq * O;
  maxpool_kernel<<<(npool + 255) / 256, 256, 0, stream>>>(feat, ws + O_POOL,
                                                          nq, NSAMPLE, O);
  size_t nc = (size_t)nq * (3 + O);
  concat_kernel<<<(int)((nc + 255) / 256), 256, 0, stream>>>(
      sel, Cin, ws + O_POOL, O, stageOut, nc);
}

static void fp_stage(hipStream_t stream, void* const* d_in, const float* p1,
                     int N1, int C1, const float* p2, int N2, int C2, int pbase,
                     const int* outs, int nl, float* ws, float* outBuf) {
  int nq = BATCH * N1;
  float* IP = ws + O_IP;
  interp_kernel<<<(nq + 127) / 128, 128, 0, stream>>>(p1, C1, N1, p2, C2, N2,
                                                      IP, nq);
  int Cip = C1 + C2 - 3;
  const float* feat;
  int O;
  pointnet_run(stream, d_in, pbase, nl, outs, IP, Cip, 3, nq, Cip - 3,
               ws + O_T1, ws + O_T2, ws + O_SC, ws + O_SH, &feat, &O);
  size_t nc = (size_t)nq * (3 + O);
  concat_kernel<<<(int)((nc + 255) / 256), 256, 0, stream>>>(IP, Cip, feat, O,
                                                             outBuf, nc);
}

extern "C" void kernel_launch(void* const* d_in, const int* in_sizes, int n_in,
                              void* d_out, int out_size, void* d_ws,
                              size_t ws_size, hipStream_t stream) {
  (void)n_in; (void)out_size; (void)ws_size;
  const float* x = (const float*)d_in[0];
  float* ws = (float*)d_ws;
  float* out = (float*)d_out;

  // Disambiguate pytree flatten order at runtime from leaf sizes:
  // jax tree-sorted order puts fp1.W0 (256*768) at index 1; insertion order
  // puts sa1.W0 (32*3 = 96) there.
  int b_fp1, b_fp2, b_fp3, b_fp4, b_head, b_sa1, b_sa2, b_sa3, b_sa4;
  if (in_sizes[1] == 256 * 768) {  // sorted keys: fp1..fp4, head, sa1..sa4
    b_fp1 = 1;  b_fp2 = 9;  b_fp3 = 17; b_fp4 = 25;
    b_head = 37;
    b_sa1 = 39; b_sa2 = 51; b_sa3 = 63; b_sa4 = 75;
  } else {  // insertion order: sa1..sa4, fp1..fp4, head
    b_sa1 = 1;  b_sa2 = 13; b_sa3 = 25; b_sa4 = 37;
    b_fp1 = 49; b_fp2 = 57; b_fp3 = 65; b_fp4 = 73;
    b_head = 81;
  }

  static const int sa1_o[3] = {32, 32, 64};
  static const int sa2_o[3] = {64, 64, 128};
  static const int sa3_o[3] = {128, 128, 256};
  static const int sa4_o[3] = {256, 256, 512};
  static const int fp12_o[2] = {256, 256};
  static const int fp3_o[2] = {256, 128};
  static const int fp4_o[3] = {128, 128, 128};

  float* S1 = ws + O_S1;  // (8,1024,67)
  float* S2 = ws + O_S2;  // (8,256,131)
  float* S3 = ws + O_S3;  // (8,64,259)
  float* S4 = ws + O_S4;  // (8,16,515)
  float* F1 = ws + O_F1;  // (8,64,259)
  float* F2 = ws + O_F2;  // (8,256,259)
  float* F3 = ws + O_F3;  // (8,1024,131)
  float* F4 = ws + O_F4;  // (8,4096,131)

  // Set-abstraction stages
  sa_stage(stream, d_in, x, NPTS, 3, 1024, 0.1f, b_sa1, sa1_o, 3, ws, S1);
  sa_stage(stream, d_in, S1, 1024, 67, 256, 0.2f, b_sa2, sa2_o, 3, ws, S2);
  sa_stage(stream, d_in, S2, 256, 131, 64, 0.4f, b_sa3, sa3_o, 3, ws, S3);
  sa_stage(stream, d_in, S3, 64, 259, 16, 0.8f, b_sa4, sa4_o, 3, ws, S4);

  // Feature-propagation stages
  fp_stage(stream, d_in, S3, 64, 259, S4, 16, 515, b_fp1, fp12_o, 2, ws, F1);
  fp_stage(stream, d_in, S2, 256, 131, F1, 64, 259, b_fp2, fp12_o, 2, ws, F2);
  fp_stage(stream, d_in, S1, 1024, 67, F2, 256, 259, b_fp3, fp3_o, 2, ws, F3);
  fp_stage(stream, d_in, x, 4096, 3, F3, 1024, 131, b_fp4, fp4_o, 3, ws, F4);

  // Outputs: pos = input xyz; logits = F4[:,:,3:131] @ Whead^T + bhead
  size_t npos = (size_t)BATCH * NPTS * 3;
  copy_kernel<<<(int)((npos + 255) / 256), 256, 0, stream>>>(x, out, npos);

  const float* Wh = (const float*)d_in[b_head];
  const float* bh = (const float*)d_in[b_head + 1];
  float* logits = out + npos;
  int rows = BATCH * NPTS;  // 32768
  launch_gemm(stream, F4, 131, 3, Wh, bh, logits, rows, 128, 16);
}